// TGAT_Layer_65171833749893
// MI455X (gfx1250) — compile-verified
//
#include <hip/hip_runtime.h>

typedef float v2f __attribute__((ext_vector_type(2)));
typedef float v8f __attribute__((ext_vector_type(8)));
typedef unsigned int u32x4 __attribute__((ext_vector_type(4)));
typedef int i32x4 __attribute__((ext_vector_type(4)));
typedef int i32x8 __attribute__((ext_vector_type(8)));

#define NN 16384
#define DD 32
#define FF 128
#define TDV 64
#define ENCS 68   // 64 + pad (68 % 64 = 4 -> conflict-free row stride)
#define KVS 132   // 128 + 4 pad (matches TDM pad: interval 128 dw, amount 4 dw)

#if __has_builtin(__builtin_amdgcn_tensor_load_to_lds) && \
    __has_builtin(__builtin_amdgcn_s_wait_tensorcnt)
#define USE_TDM 1
#else
#define USE_TDM 0
#endif

// ---------------------------------------------------------------------------
// TDM: stage a contiguous [rows x 128] f32 block into LDS with a padded
// 132-float row stride (pad_enable, interval=128 DWORDs, amount=4 DWORDs).
// 1-row-tile descriptor: tile_dim0 = nelem, data_size = 4B, type = 2.
// ---------------------------------------------------------------------------
#if USE_TDM
__device__ __forceinline__ void tdm_load_pad132(void* ldsDst,
                                                const float* gsrc,
                                                unsigned nelem) {
  const unsigned lds_off = (unsigned)(uintptr_t)ldsDst;
  const unsigned long long ga = (unsigned long long)(uintptr_t)gsrc;
  u32x4 g0;
  g0[0] = 1u;  // count=1, is_restore=0, gather off
  g0[1] = lds_off;
  g0[2] = (unsigned)(ga & 0xFFFFFFFFu);
  g0[3] = (unsigned)((ga >> 32) & 0x1FFFFFFu) | (2u << 30);  // addr hi, type=2
  i32x8 g1;
  // data_size=4B (2), pad_enable, pad_interval=128dw (6), pad_amount=4dw (3)
  g1[0] = (int)((2u << 16) | (1u << 20) | (6u << 22) | (3u << 25));
  g1[1] = (int)(nelem << 16);                              // tensor_dim0 lo16
  g1[2] = (int)(((nelem >> 16) & 0xFFFFu) | (1u << 16));   // dim0 hi16, dim1=1
  g1[3] = (int)((nelem & 0xFFFFu) << 16);                  // tile_dim0
  g1[4] = 0;                                               // tile_dim1/2 unused
  g1[5] = (int)nelem;                                      // dim0 stride lo32
  g1[6] = 0;
  g1[7] = 0;
  i32x4 z4 = {0, 0, 0, 0};
#if defined(__clang_major__) && __clang_major__ >= 23
  i32x8 z8 = {};
  __builtin_amdgcn_tensor_load_to_lds(g0, g1, z4, z4, z8, 0);
#else
  __builtin_amdgcn_tensor_load_to_lds(g0, g1, z4, z4, 0);
#endif
}
#endif

// Fallback cooperative copy into the same padded layout.
__device__ __forceinline__ void copy_pad132(float* ldsDst, const float* gsrc,
                                            int nelem, int tid, int nthreads) {
  for (int i = tid; i < nelem; i += nthreads)
    ldsDst[(i >> 7) * KVS + (i & 127)] = gsrc[i];
}

// ---------------------------------------------------------------------------
// Kernel 0: tq = Time2Vec(t) @ Wtq + btq   (128 outputs, one tiny block)
// ---------------------------------------------------------------------------
__global__ __launch_bounds__(128) void t2v_tq_kernel(
    const float* __restrict__ t, const float* __restrict__ w0,
    const float* __restrict__ b0, const float* __restrict__ Wt,
    const float* __restrict__ Bt, const float* __restrict__ Wtq,
    const float* __restrict__ btq, float* __restrict__ tq) {
  __shared__ float enc[TDV];
  const int tid = threadIdx.x;
  const float tv = t[0];
  if (tid < TDV) {
    enc[tid] = (tid == 0) ? (tv * w0[0] + b0[0])
                          : __sinf(tv * Wt[tid - 1] + Bt[tid - 1]);
  }
  __syncthreads();
  float acc = btq[tid];
  for (int j = 0; j < TDV; ++j) acc += enc[j] * Wtq[j * FF + tid];
  tq[tid] = acc;
}

// ---------------------------------------------------------------------------
// Kernel 1: k_/q/v_ = x @ {Wk,Wq,Wv} + bias  (q bias folded with tq)
// grid = (N/16, 3), block = 256 (8 waves), wave w -> 16-col tile w.
// Weight staged once per block via TDM into padded LDS; B fed from ds reads.
// ---------------------------------------------------------------------------
__global__ __launch_bounds__(256) void kqv_kernel(
    const float* __restrict__ x,
    const float* __restrict__ Wk, const float* __restrict__ bk,
    const float* __restrict__ Wq, const float* __restrict__ bq,
    const float* __restrict__ Wv, const float* __restrict__ bv,
    const float* __restrict__ tq,
    float* __restrict__ ok, float* __restrict__ oq, float* __restrict__ ov) {
  __shared__ float Wlds[FF * KVS];  // 128 rows, padded stride 132
  const int rowBase = blockIdx.x * 16;
  const int which = blockIdx.y;
  const float* W;
  const float* bias;
  float* out;
  if (which == 0)      { W = Wk; bias = bk; out = ok; }
  else if (which == 1) { W = Wq; bias = bq; out = oq; }
  else                 { W = Wv; bias = bv; out = ov; }

#if USE_TDM
  if (threadIdx.x < 32) {
    tdm_load_pad132(Wlds, W, FF * FF);
    __builtin_amdgcn_s_wait_tensorcnt(0);
  }
#else
  copy_pad132(Wlds, W, FF * FF, threadIdx.x, 256);
#endif
  __syncthreads();

  const int lane   = threadIdx.x & 31;
  const int wave   = threadIdx.x >> 5;
  const int laneLo = lane & 15;
  const int koff   = (lane >> 4) << 1;   // 0 or 2 (A/B K-pair select)
  const int mOff   = (lane >> 4) * 8;    // C/D: lanes 16-31 hold M=8..15
  const int col    = wave * 16 + laneLo;
  const int arow   = rowBase + laneLo;

  v8f acc = {};
  for (int kk = 0; kk < FF; kk += 4) {
    const int k0 = kk + koff;
    v2f a, b;
    a.x = x[arow * FF + k0];
    a.y = x[arow * FF + k0 + 1];
    b.x = Wlds[k0 * KVS + col];
    b.y = Wlds[(k0 + 1) * KVS + col];
    acc = __builtin_amdgcn_wmma_f32_16x16x4_f32(false, a, false, b,
                                                (short)0, acc, false, false);
  }
  float bcol = bias[col];
  if (which == 1) bcol += tq[col];
  for (int i = 0; i < 8; ++i)
    out[(rowBase + mOff + i) * FF + col] = acc[i] + bcol;
}

// ---------------------------------------------------------------------------
// Kernel 2: per-node fused Time2Vec -> tk/tv GEMM (WMMA, A+B from LDS) ->
// gather-add k_/v_ -> masked multi-head attention -> h[n,128]
// TDM of Wtk/Wtv issued at entry and overlapped with the sin-heavy T2V phase.
// grid = N, block = 256 (8 waves). Wave = head for the attention phase.
// ---------------------------------------------------------------------------
__global__ __launch_bounds__(256) void attn_kernel(
    const int* __restrict__ neighbors, const float* __restrict__ times,
    const float* __restrict__ t,
    const float* __restrict__ w0, const float* __restrict__ b0,
    const float* __restrict__ Wt, const float* __restrict__ Bt,
    const float* __restrict__ Wtk, const float* __restrict__ btk,
    const float* __restrict__ Wtv, const float* __restrict__ btv,
    const float* __restrict__ kmat, const float* __restrict__ qmat,
    const float* __restrict__ vmat, float* __restrict__ hout) {
  __shared__ float Wlds2[2][TDV * KVS];  // staged Wtk / Wtv, padded rows
  __shared__ float enc[DD][ENCS];
  __shared__ float kbuf[DD][KVS];
  __shared__ float vbuf[DD][KVS];
  __shared__ int   nbrS[DD];
  __shared__ float tS[DD];
  __shared__ float qS[FF];
  __shared__ float attnS[8][DD];

  const int n    = blockIdx.x;
  const int tid  = threadIdx.x;
  const int lane = tid & 31;
  const int wave = tid >> 5;

#if USE_TDM
  if (tid < 32) {  // wave 0 issues both DMAs; they overlap the T2V phase
    tdm_load_pad132(&Wlds2[0][0], Wtk, TDV * FF);
    tdm_load_pad132(&Wlds2[1][0], Wtv, TDV * FF);
  }
#else
  copy_pad132(&Wlds2[0][0], Wtk, TDV * FF, tid, 256);
  copy_pad132(&Wlds2[1][0], Wtv, TDV * FF, tid, 256);
#endif

  if (tid < DD) {
    nbrS[tid] = neighbors[n * DD + tid];
    tS[tid]   = times[n * DD + tid];
  }
  if (tid < FF) qS[tid] = qmat[n * FF + tid];
  __syncthreads();

  // Time2Vec encodings for 32 neighbors: 32x64 values, 8 per thread.
  const float w0v = w0[0], b0v = b0[0];
  for (int i = tid; i < DD * TDV; i += 256) {
    const int d = i >> 6, j = i & 63;
    const float ts = tS[d];
    enc[d][j] = (j == 0) ? (ts * w0v + b0v)
                         : __sinf(ts * Wt[j - 1] + Bt[j - 1]);
  }
#if USE_TDM
  if (tid < 32) __builtin_amdgcn_s_wait_tensorcnt(0);
#endif
  __syncthreads();  // enc visible + TDM weights visible

  // tk/tv = enc(32x64) @ Wtk/Wtv(64x128) + bias + gathered k_/v_ row.
  // 32 (mat x mtile x ctile) 16x16 tiles, 4 per wave. A and B from LDS.
  const int laneLo = lane & 15;
  const int koff   = (lane >> 4) << 1;
  const int mOff   = (lane >> 4) * 8;
  for (int q4 = 0; q4 < 4; ++q4) {
    const int tile  = wave * 4 + q4;
    const int mat   = tile >> 4;         // 0: K path, 1: V path
    const int mtile = (tile >> 3) & 1;   // neighbor rows 0-15 / 16-31
    const int ctile = tile & 7;
    const float* WmL = &Wlds2[mat][0];
    const float* bm  = mat ? btv : btk;
    const float* Xm  = mat ? vmat : kmat;
    float* dst       = mat ? &vbuf[0][0] : &kbuf[0][0];
    const int col    = ctile * 16 + laneLo;
    const int arow   = mtile * 16 + laneLo;

    v8f acc = {};
    for (int kk = 0; kk < TDV; kk += 4) {
      const int k0 = kk + koff;
      v2f a, b;
      a.x = enc[arow][k0];
      a.y = enc[arow][k0 + 1];
      b.x = WmL[k0 * KVS + col];
      b.y = WmL[(k0 + 1) * KVS + col];
      acc = __builtin_amdgcn_wmma_f32_16x16x4_f32(false, a, false, b,
                                                  (short)0, acc, false, false);
    }
    const float bcol = bm[col];
    for (int i = 0; i < 8; ++i) {
      const int d = mtile * 16 + mOff + i;
      dst[d * KVS + col] = acc[i] + bcol + Xm[nbrS[d] * FF + col];
    }
  }
  __syncthreads();

  // Scores + softmax: wave = head h, lane = neighbor d (wave32 reduce).
  const float t0 = t[0];
  {
    const int h = wave, d = lane;
    float s = 0.f;
    for (int j = 0; j < 16; ++j) s += qS[h * 16 + j] * kbuf[d][h * 16 + j];
    s *= 0.25f;  // 1/sqrt(HD=16)
    const bool valid = (tS[d] <= t0);
    float sm = valid ? s : -1e30f;
    float m = sm;
    for (int off = 16; off > 0; off >>= 1)
      m = fmaxf(m, __shfl_xor(m, off, 32));
    const float e = valid ? __expf(sm - m) : 0.f;  // masked lanes contribute 0
    float ssum = e;
    for (int off = 16; off > 0; off >>= 1) ssum += __shfl_xor(ssum, off, 32);
    attnS[h][d] = (ssum > 0.f) ? (e / ssum) : 0.f;  // all-masked node -> h=0
  }
  __syncthreads();

  // h[n, h*16+j] = sum_d attn[h][d] * vbuf[d][h*16+j]; lanes 0..15 = j.
  if (lane < 16) {
    const int h = wave, j = lane;
    float acc = 0.f;
    for (int d = 0; d < DD; ++d) acc += attnS[h][d] * vbuf[d][h * 16 + j];
    hout[n * FF + h * 16 + j] = acc;
  }
}

// ---------------------------------------------------------------------------
// Kernel 3: out = relu([x||h] @ W1 + b1) @ W2 + b2
// grid = N/16, block = 256 (8 waves). Two WMMA GEMMs, ReLU staged in LDS.
// ---------------------------------------------------------------------------
__global__ __launch_bounds__(256) void mlp_kernel(
    const float* __restrict__ x, const float* __restrict__ h,
    const float* __restrict__ W1, const float* __restrict__ b1,
    const float* __restrict__ W2, const float* __restrict__ b2,
    float* __restrict__ out) {
  __shared__ float abuf[16][KVS];
  const int rowBase = blockIdx.x * 16;
  const int lane    = threadIdx.x & 31;
  const int wave    = threadIdx.x >> 5;
  const int laneLo  = lane & 15;
  const int koff    = (lane >> 4) << 1;
  const int mOff    = (lane >> 4) * 8;
  const int col     = wave * 16 + laneLo;
  const int arow    = rowBase + laneLo;

  // GEMM1: K = 256, A = [x || h] rows
  v8f acc = {};
  for (int kk = 0; kk < 2 * FF; kk += 4) {
    const int k0 = kk + koff;  // even, never straddles the 128 boundary
    v2f a, b;
    if (k0 < FF) {
      a.x = x[arow * FF + k0];
      a.y = x[arow * FF + k0 + 1];
    } else {
      a.x = h[arow * FF + k0 - FF];
      a.y = h[arow * FF + k0 + 1 - FF];
    }
    b.x = W1[k0 * FF + col];
    b.y = W1[(k0 + 1) * FF + col];
    acc = __builtin_amdgcn_wmma_f32_16x16x4_f32(false, a, false, b,
                                                (short)0, acc, false, false);
  }
  const float bcol = b1[col];
  for (int i = 0; i < 8; ++i)
    abuf[mOff + i][col] = fmaxf(acc[i] + bcol, 0.f);
  __syncthreads();

  // GEMM2: K = 128, A from LDS
  v8f acc2 = {};
  for (int kk = 0; kk < FF; kk += 4) {
    const int k0 = kk + koff;
    v2f a, b;
    a.x = abuf[laneLo][k0];
    a.y = abuf[laneLo][k0 + 1];
    b.x = W2[k0 * FF + col];
    b.y = W2[(k0 + 1) * FF + col];
    acc2 = __builtin_amdgcn_wmma_f32_16x16x4_f32(false, a, false, b,
                                                 (short)0, acc2, false, false);
  }
  const float b2c = b2[col];
  for (int i = 0; i < 8; ++i)
    out[(rowBase + mOff + i) * FF + col] = acc2[i] + b2c;
}

// ---------------------------------------------------------------------------
extern "C" void kernel_launch(void* const* d_in, const int* in_sizes, int n_in,
                              void* d_out, int out_size, void* d_ws,
                              size_t ws_size, hipStream_t stream) {
  (void)in_sizes; (void)n_in; (void)out_size; (void)ws_size;
  const float* x         = (const float*)d_in[0];
  const int*   neighbors = (const int*)d_in[1];
  const float* times     = (const float*)d_in[2];
  const float* t         = (const float*)d_in[3];
  const float* Wk  = (const float*)d_in[4];
  const float* bk  = (const float*)d_in[5];
  const float* Wq  = (const float*)d_in[6];
  const float* bq  = (const float*)d_in[7];
  const float* Wv  = (const float*)d_in[8];
  const float* bv  = (const float*)d_in[9];
  const float* w0  = (const float*)d_in[10];
  const float* b0  = (const float*)d_in[11];
  const float* Wt  = (const float*)d_in[12];
  const float* Bt  = (const float*)d_in[13];
  const float* Wtk = (const float*)d_in[14];
  const float* btk = (const float*)d_in[15];
  const float* Wtq = (const float*)d_in[16];
  const float* btq = (const float*)d_in[17];
  const float* Wtv = (const float*)d_in[18];
  const float* btv = (const float*)d_in[19];
  const float* W1  = (const float*)d_in[20];
  const float* b1  = (const float*)d_in[21];
  const float* W2  = (const float*)d_in[22];
  const float* b2  = (const float*)d_in[23];

  float* ws   = (float*)d_ws;
  float* tq   = ws;                       // 128
  float* kmat = tq + FF;                  // N*128
  float* qmat = kmat + (size_t)NN * FF;   // N*128
  float* vmat = qmat + (size_t)NN * FF;   // N*128
  float* hbuf = vmat + (size_t)NN * FF;   // N*128

  t2v_tq_kernel<<<1, 128, 0, stream>>>(t, w0, b0, Wt, Bt, Wtq, btq, tq);

  kqv_kernel<<<dim3(NN / 16, 3), 256, 0, stream>>>(
      x, Wk, bk, Wq, bq, Wv, bv, tq, kmat, qmat, vmat);

  attn_kernel<<<NN, 256, 0, stream>>>(
      neighbors, times, t, w0, b0, Wt, Bt, Wtk, btk, Wtv, btv,
      kmat, qmat, vmat, hbuf);

  mlp_kernel<<<NN / 16, 256, 0, stream>>>(
      x, hbuf, W1, b1, W2, b2, (float*)d_out);
}